// FMCTR_65695819759980
// MI455X (gfx1250) — compile-verified
//
#include <hip/hip_runtime.h>
#include <hip/hip_bf16.h>

// ---------------------------------------------------------------------------
// FM-CTR (DeepFM second-order term), MI455X / gfx1250.
//
// Kernel 1: dense_embed = dense_x(16384x13) @ dense_w.T(13x128) + dense_b
//           via V_WMMA_F32_16X16X4_F32 (fp32 WMMA, exact), one 16x16 tile
//           per wave, K=13 padded to 16 -> 4 WMMA steps. Bias folded into C.
//           Output -> d_ws (16384*128*4 = 8 MB, L2-resident).
//
// Kernel 2: one wave32 per batch row. Each lane owns 4 of 128 dims (float4).
//           26 independent 512B/wave gathers (global_load_b128 per lane) for
//           MLP, plus the dense row from d_ws. Accumulate sum & sum-of-sq,
//           FM scalar via wave32 shfl_xor reduction.
// ---------------------------------------------------------------------------

#define N_TABLES  26
#define VOCAB     50000
#define EMBED_DIM 128
#define DENSE_DIM 13
#define BATCH     16384

typedef __attribute__((ext_vector_type(2))) float v2f;
typedef __attribute__((ext_vector_type(8))) float v8f;

// ---- Kernel 1: WMMA dense projection ---------------------------------------
// Grid: 1024 blocks x 256 threads (8 waves). Block b -> rows 16b..16b+15,
// wave w -> embed cols 16w..16w+15. 1024x8 = 8192 tiles covering 16384x128.
__global__ __launch_bounds__(256)
void fm_dense_embed_wmma(const float* __restrict__ dense_x,   // [BATCH,13]
                         const float* __restrict__ dense_w,   // [128,13]
                         const float* __restrict__ dense_b,   // [128]
                         float* __restrict__ dense_embed) {   // [BATCH,128]
    const int lane = threadIdx.x & 31;
    const int wave = threadIdx.x >> 5;        // 0..7 -> column tile
    const int r0   = blockIdx.x * 16;         // batch-row tile base
    const int c0   = wave * 16;               // embed-col tile base
    const int half = lane >> 4;               // 0: lanes 0-15, 1: lanes 16-31
    const int mn   = lane & 15;               // A-row / B-col / D-col index

    // C/D layout: VGPR g holds row (g + 8*half), col mn. Bias is per-column,
    // so initialize every accumulator VGPR with dense_b[c0+mn].
    const float bias = dense_b[c0 + mn];
    v8f c;
#pragma unroll
    for (int g = 0; g < 8; ++g) c[g] = bias;

    // A (16x4 f32): lane half=0 holds K=k0,k0+1; half=1 holds K=k0+2,k0+3.
    // B (4x16 f32): mirrored, N = mn per lane.
    // dense_embed[b][d] = sum_k dense_x[b][k] * dense_w[d][k]  (B[k][n]=w[c0+n][k])
#pragma unroll
    for (int kk = 0; kk < 4; ++kk) {
        const int k0 = kk * 4 + half * 2;     // global K of this lane's pair
        v2f a, b;
        a[0] = (k0     < DENSE_DIM) ? dense_x[(r0 + mn) * DENSE_DIM + k0    ] : 0.0f;
        a[1] = (k0 + 1 < DENSE_DIM) ? dense_x[(r0 + mn) * DENSE_DIM + k0 + 1] : 0.0f;
        b[0] = (k0     < DENSE_DIM) ? dense_w[(c0 + mn) * DENSE_DIM + k0    ] : 0.0f;
        b[1] = (k0 + 1 < DENSE_DIM) ? dense_w[(c0 + mn) * DENSE_DIM + k0 + 1] : 0.0f;
        // 8 args: (neg_a, A, neg_b, B, c_mod, C, reuse_a, reuse_b)
        c = __builtin_amdgcn_wmma_f32_16x16x4_f32(false, a, false, b,
                                                  (short)0, c, false, false);
    }

#pragma unroll
    for (int g = 0; g < 8; ++g)
        dense_embed[(size_t)(r0 + half * 8 + g) * EMBED_DIM + c0 + mn] = c[g];
}

// ---- Kernel 2: gather + FM reduction ---------------------------------------
// Grid: 2048 blocks x 256 threads = 16384 waves, one wave per batch row.
__global__ __launch_bounds__(256)
void fm_gather_reduce(const int* __restrict__ discrete_x,     // [BATCH,26] int32
                      const float* __restrict__ emb_tables,   // [26,VOCAB,128]
                      const float* __restrict__ dense_embed,  // [BATCH,128]
                      float* __restrict__ out) {              // [BATCH]
    const int lane = threadIdx.x & 31;
    const int row  = (int)((blockIdx.x * blockDim.x + threadIdx.x) >> 5);
    const int d4   = lane * 4;                // this lane's 4 dims

    // Start accumulators with the dense embedding row (the 27th "table").
    float4 s  = *(const float4*)(dense_embed + (size_t)row * EMBED_DIM + d4);
    float4 ss = make_float4(s.x * s.x, s.y * s.y, s.z * s.z, s.w * s.w);

    const int* __restrict__ idxp = discrete_x + row * N_TABLES;

    // 26 independent 512B-per-wave gathers; full unroll lets the compiler
    // batch global_load_b128 issues ahead of the waits (MLP for HBM).
#pragma unroll
    for (int t = 0; t < N_TABLES; ++t) {
        const unsigned idx = (unsigned)idxp[t];
        const size_t off = (((size_t)t * VOCAB + idx) * EMBED_DIM) + d4;
        const float4 e = *(const float4*)(emb_tables + off);
        s.x += e.x; s.y += e.y; s.z += e.z; s.w += e.w;
        ss.x = fmaf(e.x, e.x, ss.x);
        ss.y = fmaf(e.y, e.y, ss.y);
        ss.z = fmaf(e.z, e.z, ss.z);
        ss.w = fmaf(e.w, e.w, ss.w);
    }

    // Per-lane partial of sum_d (sum_d^2 - sumsq_d) over this lane's 4 dims.
    float p = (s.x * s.x - ss.x) + (s.y * s.y - ss.y)
            + (s.z * s.z - ss.z) + (s.w * s.w - ss.w);

    // Wave32 butterfly reduction.
#pragma unroll
    for (int o = 16; o >= 1; o >>= 1)
        p += __shfl_xor(p, o, 32);

    if (lane == 0) out[row] = 0.5f * p;
}

// ---------------------------------------------------------------------------
extern "C" void kernel_launch(void* const* d_in, const int* in_sizes, int n_in,
                              void* d_out, int out_size, void* d_ws, size_t ws_size,
                              hipStream_t stream) {
    const float* dense_x    = (const float*)d_in[0];   // [16384,13]
    const int*   discrete_x = (const int*)  d_in[1];   // [16384,26]
    const float* emb_tables = (const float*)d_in[2];   // [26,50000,128]
    const float* dense_w    = (const float*)d_in[3];   // [128,13]
    const float* dense_b    = (const float*)d_in[4];   // [128]
    float*       out        = (float*)d_out;           // [16384]
    float*       dense_emb  = (float*)d_ws;            // 16384*128*4 = 8 MB scratch

    // Kernel 1: 1024 row-tiles x 8 col-tiles (one per wave of a 256-thr block).
    fm_dense_embed_wmma<<<BATCH / 16, 256, 0, stream>>>(dense_x, dense_w,
                                                        dense_b, dense_emb);
    // Kernel 2: one wave per row; 8 waves per 256-thread block.
    fm_gather_reduce<<<BATCH / 8, 256, 0, stream>>>(discrete_x, emb_tables,
                                                    dense_emb, out);
}